// ClebschGordanLayer_36395552866961
// MI455X (gfx1250) — compile-verified
//
#include <hip/hip_runtime.h>

// CDNA5 / gfx1250: wave32, WMMA f32 16x16x4 (A,B = 2 VGPRs; C/D = 8 VGPRs)
typedef __attribute__((ext_vector_type(2))) float v2f;
typedef __attribute__((ext_vector_type(8))) float v8f;

// residual_x (15x7) = [Kron(D1_x,D2_x) (15x15) | CG (15x7)] * [CG (15x7) ; -D3_x (7x7)]
// K = 22 padded to 24 -> 6 chained V_WMMA_F32_16X16X4_F32 per batch element.
// All pad rows/cols are built as exact zeros, so the residual fragment needs no masking.

__global__ void cg_zero_out(float* out) {
    if (threadIdx.x == 0 && blockIdx.x == 0) out[0] = 0.0f;
}

__device__ __forceinline__ v8f wmma_acc(v2f a, v2f b, v8f c) {
    return __builtin_amdgcn_wmma_f32_16x16x4_f32(false, a, false, b, (short)0, c,
                                                 false, false);
}

__global__ __launch_bounds__(256)
void cg_residual_wmma_kernel(const float* __restrict__ D1,
                             const float* __restrict__ D2,
                             const float* __restrict__ D3,
                             const float* __restrict__ cg,
                             float* __restrict__ out,
                             int batch, float inv_count)
{
    __shared__ float cg_pad[16 * 16];   // zero-padded 15x7 CG matrix
    __shared__ float stage[8][128];     // per-wave: D1(9)|D2(25)|D3(49) + pad/dump area
    __shared__ float s_sum;

    const int tid         = threadIdx.x;
    const int lane        = tid & 31;
    const int waveInBlk   = tid >> 5;
    const int wavesPerBlk = blockDim.x >> 5;
    const int gwave       = blockIdx.x * wavesPerBlk + waveInBlk;
    const int totalWaves  = gridDim.x * wavesPerBlk;

    // one-time shared init: zero-padded CG tile + block accumulator
    if (tid < 256) cg_pad[tid] = 0.0f;
    if (tid == 0)  s_sum = 0.0f;
    __syncthreads();
    if (tid < 105) cg_pad[(tid / 7) * 16 + (tid % 7)] = cg[tid];
    __syncthreads();

    // WMMA fragment coordinates for this lane
    const int h = lane >> 4;              // K-half within a 4-wide tile
    const int p = lane & 15;              // A row (M) / B column (N)
    const int m = p;
    const int aRow = (m < 15) ? m / 5 : 0;
    const int bRow = (m < 15) ? m % 5 : 0;
    const float vm = (m < 15) ? 1.0f : 0.0f;   // zero A row 15
    const bool  pv = (p < 7);                  // valid B column

    // ---- loop-invariant CG fragments (hoisted out of the streaming loop) ----
    v2f bv0, bv1, bv2, bv3, av4, av5;
    bv0.x = cg_pad[( 0 + 2*h) * 16 + p];  bv0.y = cg_pad[( 1 + 2*h) * 16 + p];
    bv1.x = cg_pad[( 4 + 2*h) * 16 + p];  bv1.y = cg_pad[( 5 + 2*h) * 16 + p];
    bv2.x = cg_pad[( 8 + 2*h) * 16 + p];  bv2.y = cg_pad[( 9 + 2*h) * 16 + p];
    bv3.x = cg_pad[(12 + 2*h) * 16 + p];  bv3.y = cg_pad[(13 + 2*h) * 16 + p];
    av4.x = cg_pad[m * 16 + 2*h];         av4.y = cg_pad[m * 16 + 2*h + 1];
    av5.x = cg_pad[m * 16 + 4 + 2*h];     av5.y = cg_pad[m * 16 + 5 + 2*h]; // col7 = 0

    // ---- per-lane staging descriptors (q = lane, lane+32, lane+64 of 83) ----
    const int q0 = lane, q1 = lane + 32, q2 = lane + 64;
    const float* sp0 = (q0 < 9)  ? D1 : D2;
    const int    st0 = (q0 < 9)  ? 9  : 25;
    const int    of0 = (q0 < 9)  ? q0 : q0 - 9;
    const float* sp1 = (q1 < 34) ? D2 : D3;
    const int    st1 = (q1 < 34) ? 25 : 49;
    const int    of1 = (q1 < 34) ? q1 - 9 : q1 - 34;
    const bool   ld2 = (q2 < 83);
    const int    of2 = ld2 ? (q2 - 34) : 0;     // clamp inactive lanes in-bounds
    const int    q2d = ld2 ? q2 : (96 + lane);  // inactive lanes dump into pad area

    // loop-invariant LDS offsets of the -D3 B fragments (rows i' = k-16)
    const int  o4x = 34 + (2*h) * 7 + p;
    const int  o4y = o4x + 7;
    const int  o5x = 34 + (4 + 2*h) * 7 + p;
    const bool b5y = pv && (h == 0);            // i' = 5+2h: row 7 is pad for h=1
    const int  o5y = b5y ? (o5x + 7) : o5x;     // clamp to stay in-slice

    // running per-lane stream pointers (strength-reduced addressing)
    const float* ptr0 = sp0 + (size_t)gwave * st0 + of0;
    const float* ptr1 = sp1 + (size_t)gwave * st1 + of1;
    const float* ptr2 = D3  + (size_t)gwave * 49 + of2;
    const size_t inc0 = (size_t)st0 * totalWaves;
    const size_t inc1 = (size_t)st1 * totalWaves;
    const size_t inc2 = (size_t)49  * totalWaves;

    const int myIters = (gwave < batch) ? ((batch - 1 - gwave) / totalWaves + 1) : 0;

    float acc = 0.0f;
    float* sd = stage[waveInBlk];

    for (int it = 0; it < myIters; ++it) {
        // prefetch next iteration's 83 floats; speculative -> safe past the end
        __builtin_prefetch(ptr0 + inc0, 0, 3);
        __builtin_prefetch(ptr1 + inc1, 0, 3);
        __builtin_prefetch(ptr2 + inc2, 0, 3);

        // branch-free coalesced stage: all loads issue before stores
        const float l0 = *ptr0;
        const float l1 = *ptr1;
        const float l2 = *ptr2;
        ptr0 += inc0; ptr1 += inc1; ptr2 += inc2;
        sd[q0]  = l0;
        sd[q1]  = l1;
        sd[q2d] = l2;
        __builtin_amdgcn_wave_barrier();   // intra-wave LDS is in-order; fence scheduler

        // Kronecker A fragments: A[m, i*5+j] = D1[a,i] * D2[b,j]
        const float d10 = vm * sd[aRow * 3 + 0];
        const float d11 = vm * sd[aRow * 3 + 1];
        const float d12 = vm * sd[aRow * 3 + 2];
        const float d20 = sd[9 + bRow * 5 + 0];
        const float d21 = sd[9 + bRow * 5 + 1];
        const float d22 = sd[9 + bRow * 5 + 2];
        const float d23 = sd[9 + bRow * 5 + 3];
        const float d24 = sd[9 + bRow * 5 + 4];
        v2f av0, av1, av2, av3;
        if (h == 0) {               // K = {0,1, 4,5, 8,9, 12,13}
            av0.x = d10 * d20;  av0.y = d10 * d21;
            av1.x = d10 * d24;  av1.y = d11 * d20;
            av2.x = d11 * d23;  av2.y = d11 * d24;
            av3.x = d12 * d22;  av3.y = d12 * d23;
        } else {                    // K = {2,3, 6,7, 10,11, 14,15(pad)}
            av0.x = d10 * d22;  av0.y = d10 * d23;
            av1.x = d11 * d21;  av1.y = d11 * d22;
            av2.x = d12 * d20;  av2.y = d12 * d21;
            av3.x = d12 * d24;  av3.y = 0.0f;
        }

        // B fragments for K-tiles 4..5: -D3[i', c]
        v2f bv4, bv5;
        bv4.x = pv  ? -sd[o4x] : 0.0f;
        bv4.y = pv  ? -sd[o4y] : 0.0f;
        bv5.x = pv  ? -sd[o5x] : 0.0f;
        bv5.y = b5y ? -sd[o5y] : 0.0f;

        // residual = Kron*CG - CG*D3, as one chained 16x16x24 WMMA GEMM
        v8f c = {0.f, 0.f, 0.f, 0.f, 0.f, 0.f, 0.f, 0.f};
        c = wmma_acc(av0, bv0, c);
        c = wmma_acc(av1, bv1, c);
        c = wmma_acc(av2, bv2, c);
        c = wmma_acc(av3, bv3, c);
        c = wmma_acc(av4, bv4, c);
        c = wmma_acc(av5, bv5, c);

        // all pad entries of c are exactly 0 -> unmasked square-accumulate
        #pragma unroll
        for (int v = 0; v < 8; ++v) acc += c[v] * c[v];
    }

    atomicAdd(&s_sum, acc);
    __syncthreads();
    if (tid == 0) atomicAdd(out, s_sum * inv_count);
}

extern "C" void kernel_launch(void* const* d_in, const int* in_sizes, int n_in,
                              void* d_out, int out_size, void* d_ws, size_t ws_size,
                              hipStream_t stream) {
    const float* D1 = (const float*)d_in[0];
    const float* D2 = (const float*)d_in[1];
    const float* D3 = (const float*)d_in[2];
    const float* cg = (const float*)d_in[3];
    float* out = (float*)d_out;

    const int batch = in_sizes[0] / 9;              // D1 is [batch,3,3]
    const float inv_count = 1.0f / ((float)batch * 105.0f);

    cg_zero_out<<<1, 1, 0, stream>>>(out);
    // 1024 blocks x 8 waves = 8192 waves -> 32 batch elements per wave at BATCH=262144
    cg_residual_wmma_kernel<<<1024, 256, 0, stream>>>(D1, D2, D3, cg, out,
                                                      batch, inv_count);
}